// MULTIVIEWNetwork_45603962749839
// MI455X (gfx1250) — compile-verified
//
#include <hip/hip_runtime.h>
#include <cmath>

// ---------------------------------------------------------------------------
// TV + Adam optimizer for MULTIVIEWNetwork reference.
//   y  : [N=16, 512, 640] = 255 * input
//   O  : [512, 640], 500 Adam steps on grad of  sum_n TV(y_n - O) + l2 reg.
// Strategy: 500 ping-pong stencil kernels (steps are sequential), each block
// stages its 18x18 halo tiles of all 16 images + O into LDS via CDNA5 async
// global->LDS copies (GLOBAL_LOAD_ASYNC_TO_LDS_B32, ASYNCcnt,
// s_wait_asynccnt), then evaluates the hand-differentiated TV gradient and
// the Adam update entirely out of LDS/registers.
// ---------------------------------------------------------------------------

#define HH 512
#define WW 640
#define MAXN 16
#define BX 16
#define BY 16
#define TLX (BX + 2)
#define TLY (BY + 2)
#define TILE_ELEMS (TLX * TLY)

#define LR_F 7e-2f
#define LAMBDA_O_F 5e-2f
#define ADAM_EPS_F 1e-8f
#define TV_EPS_F 1e-20f

#if defined(__AMDGCN__) && __has_builtin(__builtin_amdgcn_global_load_async_to_lds_b32)
#define HAVE_ASYNC_LDS 1
#else
#define HAVE_ASYNC_LDS 0
#endif

typedef __attribute__((address_space(1))) int glds_int;
typedef __attribute__((address_space(3))) int lds_int;

__device__ __forceinline__ void async_copy_b32(const float* gsrc, float* ldst) {
#if HAVE_ASYNC_LDS
    // gfx1250: GLOBAL_LOAD_ASYNC_TO_LDS_B32 (per-lane global addr -> per-lane LDS addr)
    __builtin_amdgcn_global_load_async_to_lds_b32(
        (glds_int*)gsrc, (lds_int*)ldst, /*offset=*/0, /*cpol=*/0);
#else
    *ldst = *gsrc;
#endif
}

__device__ __forceinline__ void async_wait_all() {
#if HAVE_ASYNC_LDS
#if __has_builtin(__builtin_amdgcn_s_wait_asynccnt)
    __builtin_amdgcn_s_wait_asynccnt(0);
#else
    asm volatile("s_wait_asynccnt 0" ::: "memory");
#endif
#endif
}

// Stage one (TLY x TLX) halo tile into LDS with clamped (replicated) borders.
// Pure add-based index math: ti in {ty, ty+BY}, tj in {tx, tx+BX}.
__device__ __forceinline__ void stage_tile_async(const float* __restrict__ gbase,
                                                 float* __restrict__ lbase,
                                                 int i0, int j0, int tx, int ty) {
#pragma unroll
    for (int dy = 0; dy < TLY; dy += BY) {
        const int ti = ty + dy;
        if (ti < TLY) {
            const int gi = min(max(i0 + ti, 0), HH - 1);
            const float* grow = gbase + gi * WW;
            float* lrow = lbase + ti * TLX;
#pragma unroll
            for (int dx = 0; dx < TLX; dx += BX) {
                const int tj = tx + dx;
                if (tj < TLX) {
                    const int gj = min(max(j0 + tj, 0), WW - 1);
                    async_copy_b32(grow + gj, lrow + tj);
                }
            }
        }
    }
}

// One Adam step: reads O from Oin, writes O to Oout, updates m,v in place.
__global__ __launch_bounds__(BX * BY)
void tv_adam_step(const float* __restrict__ yin, int nimg,
                  const float* __restrict__ Oin, float* __restrict__ Oout,
                  float* __restrict__ m, float* __restrict__ v,
                  float inv_bc1, float inv_bc2) {
    __shared__ float sO[TILE_ELEMS];
    __shared__ float sY[MAXN][TILE_ELEMS];

    const int tx = threadIdx.x, ty = threadIdx.y;
    const int i0 = (int)blockIdx.y * BY - 1;   // tile origin (with halo)
    const int j0 = (int)blockIdx.x * BX - 1;

    // --- stage O tile + all image tiles via async global->LDS copies --------
    stage_tile_async(Oin, sO, i0, j0, tx, ty);
    for (int n = 0; n < nimg; ++n) {
        stage_tile_async(yin + (size_t)n * HH * WW, sY[n], i0, j0, tx, ty);
    }
    async_wait_all();
    __syncthreads();

    // --- stencil gradient ---------------------------------------------------
    const int c = (ty + 1) * TLX + (tx + 1);
    const int i = blockIdx.y * BY + ty;
    const int j = blockIdx.x * BX + tx;

    const float maskI  = (i < HH - 1) ? 1.f : 0.f;  // forward-diff rows valid
    const float maskJ  = (j < WW - 1) ? 1.f : 0.f;  // forward-diff cols valid
    const float maskUp = (i > 0) ? 1.f : 0.f;       // (i-1,j) term exists
    const float maskLf = (j > 0) ? 1.f : 0.f;       // (i,j-1) term exists

    const float Oc   = sO[c];
    const float dOi  = sO[c + TLX] - Oc;            // O(i+1,j)-O(i,j)
    const float dOj  = sO[c + 1] - Oc;              // O(i,j+1)-O(i,j)
    const float dOiU = Oc - sO[c - TLX];            // gi at (i-1,j)
    const float dOjU = sO[c - TLX + 1] - sO[c - TLX];
    const float dOiL = sO[c + TLX - 1] - sO[c - 1]; // gi at (i,j-1)
    const float dOjL = Oc - sO[c - 1];

    float gsum = 0.f;
#pragma unroll 4
    for (int n = 0; n < nimg; ++n) {
        const float* t = sY[n];
        const float yc  = t[c];
        const float yD  = t[c + TLX];
        const float yR  = t[c + 1];
        const float yU  = t[c - TLX];
        const float yUR = t[c - TLX + 1];
        const float yL  = t[c - 1];
        const float yDL = t[c + TLX - 1];
        // r = 255*y_in - O; forward differences, zero past far edge
        const float giC = (255.f * (yD - yc) - dOi) * maskI;
        const float gjC = (255.f * (yR - yc) - dOj) * maskJ;
        const float giU = (255.f * (yc - yU) - dOiU);
        const float gjU = (255.f * (yUR - yU) - dOjU) * maskJ;
        const float giL = (255.f * (yDL - yL) - dOiL) * maskI;
        const float gjL = (255.f * (yc - yL) - dOjL);
        const float rC = rsqrtf(giC * giC + gjC * gjC + TV_EPS_F);
        const float rU = rsqrtf(giU * giU + gjU * gjU + TV_EPS_F);
        const float rL = rsqrtf(giL * giL + gjL * gjL + TV_EPS_F);
        gsum += (giC + gjC) * rC - maskUp * (giU * rU) - maskLf * (gjL * rL);
    }

    // --- Adam update ---------------------------------------------------------
    const float g = gsum + (2.f * LAMBDA_O_F) * Oc;
    const int idx = i * WW + j;
    const float mm = 0.9f * m[idx] + 0.1f * g;
    const float vv = 0.999f * v[idx] + 0.001f * (g * g);
    m[idx] = mm;
    v[idx] = vv;
    const float mh = mm * inv_bc1;
    const float vh = vv * inv_bc2;
    Oout[idx] = Oc - LR_F * mh / (sqrtf(vh) + ADAM_EPS_F);
}

__global__ void tv_init(float* __restrict__ O, float* __restrict__ m,
                        float* __restrict__ v) {
    const int k = blockIdx.x * blockDim.x + threadIdx.x;
    if (k < HH * WW) {
        O[k] = 0.f;
        m[k] = 0.f;
        v[k] = 0.f;
    }
}

__global__ void tv_final(const float* __restrict__ O, float* __restrict__ out) {
    const int k = blockIdx.x * blockDim.x + threadIdx.x;
    if (k < HH * WW) out[k] = O[k] * (1.0f / 255.0f);
}

extern "C" void kernel_launch(void* const* d_in, const int* in_sizes, int n_in,
                              void* d_out, int out_size, void* d_ws, size_t ws_size,
                              hipStream_t stream) {
    const float* in = (const float*)d_in[0];
    int nimg = in_sizes[0] / (HH * WW);
    if (nimg > MAXN) nimg = MAXN;
    if (nimg < 1) nimg = 1;

    float* ws = (float*)d_ws;
    const size_t P = (size_t)HH * WW;
    float* OA = ws;
    float* OB = ws + P;
    float* m  = ws + 2 * P;
    float* v  = ws + 3 * P;

    const dim3 b1d(256), g1d((HH * WW + 255) / 256);
    tv_init<<<g1d, b1d, 0, stream>>>(OA, m, v);

    const dim3 blk(BX, BY), grd(WW / BX, HH / BY);
    const float* src = OA;
    float* dst = OB;
    double p1 = 1.0, p2 = 1.0;
    for (int t = 1; t <= 500; ++t) {
        p1 *= 0.9;
        p2 *= 0.999;
        const float ib1 = (float)(1.0 / (1.0 - p1));
        const float ib2 = (float)(1.0 / (1.0 - p2));
        tv_adam_step<<<grd, blk, 0, stream>>>(in, nimg, src, dst, m, v, ib1, ib2);
        float* tmp = (float*)src;
        src = dst;
        dst = tmp;
    }
    tv_final<<<g1d, b1d, 0, stream>>>(src, (float*)d_out);
}